// PostProcessor_33097017983481
// MI455X (gfx1250) — compile-verified
//
#include <hip/hip_runtime.h>
#include <math.h>

#define BIMG 2
#define NPRO 384
#define NCLS 11
#define NFG  10
#define RCN  5
#define GTOT (BIMG*NFG*NPRO)   /* 7680 decoded fg entries */
#define FLAT (NFG*NPRO)        /* 3840 per image */
#define DETS 100
#define SCORE_T 0.05f
#define NMS_T   0.5f
#define CLIPV   4.135166556742356f   /* log(1000/16) */
#define EPSA    1e-8f

typedef float v2f __attribute__((ext_vector_type(2)));
typedef float v8f __attribute__((ext_vector_type(8)));

__device__ __forceinline__ float crossf(float ax, float ay, float bx, float by) {
    return ax * by - ay * bx;
}

// ---------------------------------------------------------------------------
// Kernel 1: softmax + rotated-box decode + geometry precompute
// ---------------------------------------------------------------------------
__global__ void pp_decode(const float* __restrict__ logits,
                          const float* __restrict__ breg,
                          const float* __restrict__ rrects,
                          float* __restrict__ dec,  float* __restrict__ scr,
                          float* __restrict__ cor,  float* __restrict__ area,
                          float* __restrict__ rad,  float* __restrict__ nsq) {
    int g = blockIdx.x * blockDim.x + threadIdx.x;
    if (g >= GTOT) return;
    int b  = g / FLAT;
    int r1 = g - b * FLAT;
    int cf = r1 / NPRO;
    int n  = r1 - cf * NPRO;
    int c  = cf + 1;
    int m  = b * NPRO + n;

    // softmax probability for class c
    float l[NCLS];
    float mx = -INFINITY;
#pragma unroll
    for (int k = 0; k < NCLS; ++k) { l[k] = logits[m * NCLS + k]; mx = fmaxf(mx, l[k]); }
    float sum = 0.f;
#pragma unroll
    for (int k = 0; k < NCLS; ++k) sum += expf(l[k] - mx);
    float prob = expf(l[c] - mx) / sum;

    // decode: weights (10,10,5,5,1), clip dw/dh above at log(1000/16)
    const float* rr = breg + (size_t)m * (NCLS * RCN) + c * RCN;
    float dx = rr[0] * 0.1f;
    float dy = rr[1] * 0.1f;
    float dw = fminf(rr[2] * 0.2f, CLIPV);
    float dh = fminf(rr[3] * 0.2f, CLIPV);
    float da = rr[4];
    const float* an = rrects + (size_t)m * RCN;
    float xc = an[0], yc = an[1], w = an[2], h = an[3], aa = an[4];
    float pxv = dx * w + xc;
    float pyv = dy * h + yc;
    float pw  = expf(dw) * w;
    float ph  = expf(dh) * h;
    float pa  = da * (180.0f / (float)M_PI) + aa;
    pa = fmodf(pa + 180.0f, 360.0f);
    if (pa < 0.f) pa += 360.0f;       // python-style mod (sign of divisor)
    pa -= 180.0f;

    float* dp = dec + (size_t)g * RCN;
    dp[0] = pxv; dp[1] = pyv; dp[2] = pw; dp[3] = ph; dp[4] = pa;
    scr[g] = prob;

    float t  = pa * ((float)M_PI / 180.0f);
    float cs = cosf(t), sn = sinf(t);
    float hx = pw * 0.5f, hy = ph * 0.5f;
    const float lx[4] = { -hx,  hx, hx, -hx };
    const float ly[4] = { -hy, -hy, hy,  hy };
    float* cp = cor + (size_t)g * 8;
#pragma unroll
    for (int k = 0; k < 4; ++k) {
        cp[2 * k]     = pxv + lx[k] * cs - ly[k] * sn;
        cp[2 * k + 1] = pyv + lx[k] * sn + ly[k] * cs;
    }
    area[g] = pw * ph;
    rad[g]  = 0.5f * sqrtf(pw * pw + ph * ph);
    nsq[g]  = pxv * pxv + pyv * pyv;
}

// ---------------------------------------------------------------------------
// Kernel 2: dense bounding-circle gate via V_WMMA_F32_16X16X4_F32
// dist^2(i,j) = |ci|^2 + |cj|^2 - 2*ci.cj ; the ci.cj matrix is a K=2 GEMM.
// A layout (ISA 7.12.2, 32-bit 16x4): lanes 0-15 regs {K=0,K=1}; lanes 16-31
// regs {K=2,K=3} (zeroed here). B mirrors with N across lanes. D: reg r holds
// M=r (lanes 0-15) / M=r+8 (lanes 16-31), N = lane%16.
// ---------------------------------------------------------------------------
__global__ void pp_gate(const float* __restrict__ dec,
                        const float* __restrict__ nsq,
                        const float* __restrict__ rad,
                        unsigned char* __restrict__ gate) {
    int bc = blockIdx.x;   // b*NFG + cf
    __shared__ float sx[NPRO], sy[NPRO], sn2[NPRO], sr[NPRO];
    int base = bc * NPRO;
    for (int i = threadIdx.x; i < NPRO; i += blockDim.x) {
        sx[i]  = dec[(size_t)(base + i) * RCN + 0];
        sy[i]  = dec[(size_t)(base + i) * RCN + 1];
        sn2[i] = nsq[base + i];
        sr[i]  = rad[base + i];
    }
    __syncthreads();

    int wave = threadIdx.x >> 5;       // wave32
    int lane = threadIdx.x & 31;
    int lid  = lane & 15;
    bool lo  = lane < 16;
    unsigned char* gbc = gate + (size_t)bc * NPRO * NPRO;

    for (int tt = wave; tt < 24 * 24; tt += (int)(blockDim.x >> 5)) {
        int ti = tt / 24, tj = tt - ti * 24;
        int i0 = ti * 16, j0 = tj * 16;
        v2f a, bb;
        a.x  = lo ? sx[i0 + lid] : 0.f;   // K=0 (or zero for K=2)
        a.y  = lo ? sy[i0 + lid] : 0.f;   // K=1 (or zero for K=3)
        bb.x = lo ? sx[j0 + lid] : 0.f;
        bb.y = lo ? sy[j0 + lid] : 0.f;
        v8f cacc = {};
        v8f d = __builtin_amdgcn_wmma_f32_16x16x4_f32(
            false, a, false, bb, (short)0, cacc, false, false);
        int   j  = j0 + lid;
        float nj = sn2[j], rj = sr[j];
#pragma unroll
        for (int r = 0; r < 8; ++r) {
            int   i     = i0 + (lo ? r : r + 8);
            float dist2 = sn2[i] + nj - 2.f * d[r];
            float rsum  = sr[i] + rj;
            // +16 absolute slack absorbs WMMA/FMA rounding (false pos = harmless)
            gbc[(size_t)i * NPRO + j] = (dist2 <= rsum * rsum + 16.f) ? 1 : 0;
        }
    }
}

// ---------------------------------------------------------------------------
// Faithful port of reference inter_area (incl. centroid-parked points and
// stable 24-point angular sort — the reference's exact, quirky formula).
// ---------------------------------------------------------------------------
__device__ float quad_inter(const float* __restrict__ ca, const float* __restrict__ cb) {
    float d1x[4], d1y[4], d2x[4], d2y[4];
#pragma unroll
    for (int i = 0; i < 4; ++i) {
        int ip = (i + 1) & 3;
        d1x[i] = ca[2 * ip] - ca[2 * i]; d1y[i] = ca[2 * ip + 1] - ca[2 * i + 1];
        d2x[i] = cb[2 * ip] - cb[2 * i]; d2y[i] = cb[2 * ip + 1] - cb[2 * i + 1];
    }
    float px[24], py[24];
    bool  mk[24];
    int idx = 0;
#pragma unroll
    for (int i = 0; i < 4; ++i)
#pragma unroll
        for (int j = 0; j < 4; ++j) {
            float den  = crossf(d1x[i], d1y[i], d2x[j], d2y[j]);
            float dfx  = cb[2 * j] - ca[2 * i];
            float dfy  = cb[2 * j + 1] - ca[2 * i + 1];
            float dens = (fabsf(den) < EPSA) ? 1.f : den;
            float t    = crossf(dfx, dfy, d2x[j], d2y[j]) / dens;
            float u    = crossf(dfx, dfy, d1x[i], d1y[i]) / dens;
            bool ok = (fabsf(den) >= EPSA) && (t >= 0.f) && (t <= 1.f) && (u >= 0.f) && (u <= 1.f);
            px[idx] = ca[2 * i] + t * d1x[i];
            py[idx] = ca[2 * i + 1] + t * d1y[i];
            mk[idx] = ok;
            ++idx;
        }
#pragma unroll
    for (int k = 0; k < 4; ++k) {     // corners of A inside B
        bool allp = true, alln = true;
#pragma unroll
        for (int j = 0; j < 4; ++j) {
            float cr = crossf(d2x[j], d2y[j], ca[2 * k] - cb[2 * j], ca[2 * k + 1] - cb[2 * j + 1]);
            allp = allp && (cr >= -1e-6f);
            alln = alln && (cr <=  1e-6f);
        }
        px[16 + k] = ca[2 * k]; py[16 + k] = ca[2 * k + 1]; mk[16 + k] = allp || alln;
    }
#pragma unroll
    for (int k = 0; k < 4; ++k) {     // corners of B inside A
        bool allp = true, alln = true;
#pragma unroll
        for (int j = 0; j < 4; ++j) {
            float cr = crossf(d1x[j], d1y[j], cb[2 * k] - ca[2 * j], cb[2 * k + 1] - ca[2 * j + 1]);
            allp = allp && (cr >= -1e-6f);
            alln = alln && (cr <=  1e-6f);
        }
        px[20 + k] = cb[2 * k]; py[20 + k] = cb[2 * k + 1]; mk[20 + k] = allp || alln;
    }
    int nv = 0; float sxs = 0.f, sys = 0.f;
#pragma unroll
    for (int k = 0; k < 24; ++k)
        if (mk[k]) { ++nv; sxs += px[k]; sys += py[k]; }
    float inv = 1.f / (float)(nv > 1 ? nv : 1);
    float cx = sxs * inv, cy = sys * inv;
    float ang[24];
#pragma unroll
    for (int k = 0; k < 24; ++k) {
        if (!mk[k]) { px[k] = cx; py[k] = cy; }   // park invalid at centroid
        ang[k] = atan2f(py[k] - cy, px[k] - cx);  // parked: atan2(0,0)=0, as jnp
    }
    for (int k = 1; k < 24; ++k) {    // stable insertion sort == stable argsort
        float a = ang[k], x = px[k], y = py[k];
        int m2 = k - 1;
        while (m2 >= 0 && ang[m2] > a) {
            ang[m2 + 1] = ang[m2]; px[m2 + 1] = px[m2]; py[m2 + 1] = py[m2];
            --m2;
        }
        ang[m2 + 1] = a; px[m2 + 1] = x; py[m2 + 1] = y;
    }
    float s = 0.f;
#pragma unroll
    for (int k = 0; k < 24; ++k) {
        int kp = (k + 1) % 24;
        s += crossf(px[k], py[k], px[kp], py[kp]);
    }
    float areav = 0.5f * fabsf(s);
    return (nv >= 3) ? areav : 0.f;
}

// ---------------------------------------------------------------------------
// Kernel 3: per-(image,class) greedy NMS, gated by the WMMA circle test
// ---------------------------------------------------------------------------
__global__ void pp_nms(const float* __restrict__ scr,
                       const float* __restrict__ cor,
                       const float* __restrict__ area,
                       const unsigned char* __restrict__ gate,
                       float* __restrict__ ks) {
    int bc = blockIdx.x;
    __shared__ float s_sc[NPRO];
    __shared__ float s_key[NPRO];
    __shared__ float s_ar[NPRO];
    __shared__ float s_co[NPRO * 8];
    __shared__ int   s_ord[NPRO];
    __shared__ unsigned char s_sup[NPRO];
    __shared__ int   s_flag;
    int base = bc * NPRO;
    for (int i = threadIdx.x; i < NPRO; i += blockDim.x) {
        float sc = scr[base + i];
        s_sc[i]  = sc;
        s_key[i] = (sc > SCORE_T) ? sc : -INFINITY;   // argsort(-masked) key
        s_ar[i]  = area[base + i];
        s_sup[i] = 0;
#pragma unroll
        for (int r = 0; r < 8; ++r) s_co[i * 8 + r] = cor[(size_t)(base + i) * 8 + r];
    }
    __syncthreads();
    // O(N^2) stable rank sort: descending key, ties by ascending index
    for (int i = threadIdx.x; i < NPRO; i += blockDim.x) {
        float ki = s_key[i];
        int rk = 0;
        for (int j = 0; j < NPRO; ++j) {
            float kj = s_key[j];
            if (kj > ki || (kj == ki && j < i)) ++rk;
        }
        s_ord[rk] = i;
    }
    __syncthreads();
    const unsigned char* gbc = gate + (size_t)bc * NPRO * NPRO;
    for (int p = 0; p < NPRO; ++p) {
        if (threadIdx.x == 0) {
            int oi = s_ord[p];
            s_flag = (!s_sup[p]) && (s_sc[oi] > SCORE_T);
        }
        __syncthreads();
        if (s_flag) {
            int oi = s_ord[p];
            float ai = s_ar[oi];
            const float* ci = &s_co[oi * 8];
            for (int q = p + 1 + threadIdx.x; q < NPRO; q += blockDim.x) {
                if (s_sup[q]) continue;
                int oj = s_ord[q];
                if (!gbc[(size_t)oi * NPRO + oj]) continue;   // circle-reject
                float inter = quad_inter(ci, &s_co[oj * 8]);
                float iou = inter / (ai + s_ar[oj] - inter + EPSA);
                if (iou > NMS_T) s_sup[q] = 1;
            }
        }
        __syncthreads();
    }
    for (int p = threadIdx.x; p < NPRO; p += blockDim.x) {
        int oi = s_ord[p];
        bool keep = (!s_sup[p]) && (s_sc[oi] > SCORE_T);
        int b = bc / NFG, cf = bc % NFG;
        ks[(size_t)b * FLAT + cf * NPRO + oi] = keep ? s_sc[oi] : -1.f;
    }
}

// ---------------------------------------------------------------------------
// Kernel 4: per-image top-100 (iterative argmax, jnp top_k tie semantics)
// ---------------------------------------------------------------------------
__global__ void pp_topk(const float* __restrict__ ks,
                        const float* __restrict__ dec,
                        float* __restrict__ out) {
    int b = blockIdx.x;
    __shared__ float sv[FLAT];
    __shared__ float rv[256];
    __shared__ int   ri[256];
    for (int i = threadIdx.x; i < FLAT; i += blockDim.x) sv[i] = ks[(size_t)b * FLAT + i];
    __syncthreads();
    for (int k = 0; k < DETS; ++k) {
        float bv = -INFINITY; int bi = 0x7fffffff;
        for (int i = threadIdx.x; i < FLAT; i += blockDim.x) {
            float v = sv[i];
            if (v > bv || (v == bv && i < bi)) { bv = v; bi = i; }
        }
        rv[threadIdx.x] = bv; ri[threadIdx.x] = bi;
        __syncthreads();
        for (int s = 128; s > 0; s >>= 1) {
            if ((int)threadIdx.x < s) {
                float ov = rv[threadIdx.x + s]; int oi = ri[threadIdx.x + s];
                float mv = rv[threadIdx.x];     int mi = ri[threadIdx.x];
                if (ov > mv || (ov == mv && oi < mi)) { rv[threadIdx.x] = ov; ri[threadIdx.x] = oi; }
            }
            __syncthreads();
        }
        if (threadIdx.x == 0) {
            float v = rv[0]; int j = ri[0];
            bool ok = v > 0.f;
            float* o = out + ((size_t)b * DETS + k) * 6;
            if (ok) {
                const float* bx = dec + ((size_t)b * FLAT + j) * RCN;
                o[0] = bx[0]; o[1] = bx[1]; o[2] = bx[2]; o[3] = bx[3]; o[4] = bx[4];
                o[5] = v;
                out[BIMG * DETS * 6 + b * DETS + k] = (float)(j / NPRO + 1);
            } else {
                o[0] = o[1] = o[2] = o[3] = o[4] = o[5] = 0.f;
                out[BIMG * DETS * 6 + b * DETS + k] = 0.f;
            }
            sv[j] = -INFINITY;   // distinct indices, like top_k
        }
        __syncthreads();
    }
}

// ---------------------------------------------------------------------------
extern "C" void kernel_launch(void* const* d_in, const int* in_sizes, int n_in,
                              void* d_out, int out_size, void* d_ws, size_t ws_size,
                              hipStream_t stream) {
    (void)in_sizes; (void)n_in; (void)out_size; (void)ws_size;
    const float* logits = (const float*)d_in[0];
    const float* breg   = (const float*)d_in[1];
    const float* rrects = (const float*)d_in[2];
    float* out = (float*)d_out;

    float* ws   = (float*)d_ws;
    float* dec  = ws;                              // GTOT*5
    float* scr  = dec  + (size_t)GTOT * RCN;       // GTOT
    float* cor  = scr  + GTOT;                     // GTOT*8
    float* area = cor  + (size_t)GTOT * 8;         // GTOT
    float* rad  = area + GTOT;                     // GTOT
    float* nsq  = rad  + GTOT;                     // GTOT
    float* ks   = nsq  + GTOT;                     // BIMG*FLAT
    unsigned char* gate = (unsigned char*)(ks + (size_t)BIMG * FLAT); // 20*384*384

    pp_decode<<<GTOT / 256, 256, 0, stream>>>(logits, breg, rrects,
                                              dec, scr, cor, area, rad, nsq);
    pp_gate<<<BIMG * NFG, 256, 0, stream>>>(dec, nsq, rad, gate);
    pp_nms<<<BIMG * NFG, 256, 0, stream>>>(scr, cor, area, gate, ks);
    pp_topk<<<BIMG, 256, 0, stream>>>(ks, dec, out);
}